// FPS_12386685681722
// MI455X (gfx1250) — compile-verified
//
#include <hip/hip_runtime.h>
#include <cstdint>

// ---- Problem geometry (fixed by the reference: B=8, N=65536, m=512) ----
#define BATCHES          8
#define NPTS             65536
#define TPB              1024
#define WAVES            32        // TPB / 32 (wave32)
#define PTS_PER_WAVE     2048      // NPTS / WAVES
#define TILE_PTS         128       // points per staged tile (1536 bytes)
#define TILES            16        // PTS_PER_WAVE / TILE_PTS
#define DIST_REGS        64        // points per lane = PTS_PER_WAVE / 32

// One tile = 128 points * 12B = 1536B = 3 x (32 lanes x 16B) async b128 copies.
__device__ __forceinline__ void async_stage_tile(uint32_t ldsByte, uint32_t gByteOff,
                                                 const float* gbase) {
  asm volatile("global_load_async_to_lds_b128 %0, %1, %2"
               :: "v"(ldsByte), "v"(gByteOff), "s"(gbase) : "memory");
  asm volatile("global_load_async_to_lds_b128 %0, %1, %2"
               :: "v"(ldsByte + 512u), "v"(gByteOff + 512u), "s"(gbase) : "memory");
  asm volatile("global_load_async_to_lds_b128 %0, %1, %2"
               :: "v"(ldsByte + 1024u), "v"(gByteOff + 1024u), "s"(gbase) : "memory");
}
__device__ __forceinline__ void wait_async_le3() {
  asm volatile("s_wait_asynccnt 3" ::: "memory");
}
__device__ __forceinline__ void wait_async_0() {
  asm volatile("s_wait_asynccnt 0" ::: "memory");
}

// argmax combine with JAX first-index tie-break
__device__ __forceinline__ void red_combine(float& d, int& i, float od, int oi) {
  if (od > d || (od == d && oi < i)) { d = od; i = oi; }
}

__global__ __launch_bounds__(TPB) void fps_kernel(const float* __restrict__ pos,
                                                  int* __restrict__ out, int m) {
  // 32 waves * 2 buffers * 1536B = 96KB staging (fits any plausible LDS cap)
  __shared__ __align__(16) float sStage[WAVES * 2 * TILE_PTS * 3];
  __shared__ float sRedD[WAVES];
  __shared__ int   sRedI[WAVES];
  __shared__ float sQ[3];
  __shared__ int   sLast;

  const int b    = blockIdx.x;
  const int tid  = threadIdx.x;
  const int w    = tid >> 5;
  const int lane = tid & 31;

  // byte offsets (fit in u32: max ~6.3MB)
  const uint32_t posByteB = (uint32_t)b * (NPTS * 12u) + (uint32_t)w * (PTS_PER_WAVE * 12u);
  const uint32_t ldsStage = (uint32_t)(uintptr_t)&sStage[0];   // low 32 bits = LDS offset
  const uint32_t ldsWave  = ldsStage + (uint32_t)w * (2u * TILE_PTS * 12u);
  const uint32_t l16      = (uint32_t)lane * 16u;

  // running min squared distance to selected set: entirely in VGPRs
  float dist[DIST_REGS];
#pragma unroll
  for (int i = 0; i < DIST_REGS; ++i) dist[i] = 3.402823466e38f;  // finfo(f32).max

  if (tid == 0) {
    sLast = 0;                                   // random_first=False -> seed index 0
    const float* q = pos + (size_t)b * (NPTS * 3);
    sQ[0] = q[0]; sQ[1] = q[1]; sQ[2] = q[2];
  }
  __syncthreads();

  for (int it = 0; it < m; ++it) {
    const float qx = sQ[0], qy = sQ[1], qz = sQ[2];
    if (tid == 0) out[b * m + it] = sLast;       // scan emits 'last' BEFORE the update

    // prefetch tile 0 into buffer 0
    async_stage_tile(ldsWave + l16, posByteB + l16, pos);

    float bestD = -1.0f;    // all candidates are >= 0
    int   bestI = 0;

#pragma unroll
    for (int t = 0; t < TILES; ++t) {
      const uint32_t cur = (uint32_t)(t & 1);
      if (t + 1 < TILES) {
        const uint32_t nb = cur ^ 1u;
        async_stage_tile(ldsWave + nb * 1536u + l16,
                         posByteB + (uint32_t)(t + 1) * 1536u + l16, pos);
        wait_async_le3();   // in-order completion: <=3 outstanding => tile t landed
      } else {
        wait_async_0();
      }

      // lane's 4 points: 48 contiguous bytes (16B aligned) in the staged AoS tile
      const float4* lp = (const float4*)&sStage[w * 768 + (int)cur * 384 + lane * 12];
      float4 a = lp[0], c4 = lp[1], e = lp[2];
      const float px[4] = {a.x, a.w, c4.z, e.y};
      const float py[4] = {a.y, c4.x, c4.w, e.z};
      const float pz[4] = {a.z, c4.y, e.x, e.w};

#pragma unroll
      for (int k = 0; k < 4; ++k) {
        const float dx = px[k] - qx, dy = py[k] - qy, dz = pz[k] - qz;
        const float dd = dx * dx + dy * dy + dz * dz;
        const int   di = t * 4 + k;
        const float nd = fminf(dist[di], dd);
        dist[di] = nd;
        const int pi = w * PTS_PER_WAVE + t * TILE_PTS + lane * 4 + k;
        if (nd > bestD) { bestD = nd; bestI = pi; }  // strict > keeps earliest index
      }
    }

    // wave32 argmax reduction (lane indices increase with lane -> tie rule exact)
#pragma unroll
    for (int off = 16; off > 0; off >>= 1) {
      float od = __shfl_xor(bestD, off, 32);
      int   oi = __shfl_xor(bestI, off, 32);
      red_combine(bestD, bestI, od, oi);
    }
    if (lane == 0) { sRedD[w] = bestD; sRedI[w] = bestI; }
    __syncthreads();

    if (w == 0) {   // 32 wave-partials map 1:1 onto wave0's 32 lanes
      float d = sRedD[lane];
      int   i = sRedI[lane];
#pragma unroll
      for (int off = 16; off > 0; off >>= 1) {
        float od = __shfl_xor(d, off, 32);
        int   oi = __shfl_xor(i, off, 32);
        red_combine(d, i, od, oi);
      }
      if (lane == 0) {
        sLast = i;
        const float* q = pos + ((size_t)b * NPTS + (size_t)i) * 3;
        sQ[0] = q[0]; sQ[1] = q[1]; sQ[2] = q[2];
      }
    }
    __syncthreads();
  }
}

extern "C" void kernel_launch(void* const* d_in, const int* in_sizes, int n_in,
                              void* d_out, int out_size, void* d_ws, size_t ws_size,
                              hipStream_t stream) {
  (void)in_sizes; (void)n_in; (void)d_ws; (void)ws_size;
  const float* pos = (const float*)d_in[0];    // [B, N, 3] f32
  int* out = (int*)d_out;                      // [B, m] int32 indices
  const int m = out_size / BATCHES;            // 512
  fps_kernel<<<dim3(BATCHES), dim3(TPB), 0, stream>>>(pos, out, m);
}